// GNNWrapper_75256416961193
// MI455X (gfx1250) — compile-verified
//
#include <hip/hip_runtime.h>
#include <math.h>

typedef float v2f __attribute__((ext_vector_type(2)));
typedef float v8f __attribute__((ext_vector_type(8)));

#define BT   16    // batch rows per block
#define NCH  128   // CH
#define NHID 256   // HID
#define OBSD 99

__global__ __launch_bounds__(256) void gnn_fused_kernel(
    const float* __restrict__ obs,      // (B,99)
    const float* __restrict__ W_fgn,    // (4,512)
    const float* __restrict__ b_fgn,    // (512,)
    const float* __restrict__ W_root,   // (4,128)
    const float* __restrict__ b_conv,   // (128,)
    const float* __restrict__ attn_k,   // (128,)
    const float* __restrict__ W_dense,  // (128,256)
    const float* __restrict__ b_dense,  // (256,)
    float* __restrict__ out)            // (B,256)
{
    __shared__ float s_obs[BT][96];          // 6 KB
    __shared__ float s_xc[BT][4][NCH];       // 32 KB
    __shared__ float s_pool[BT][132];        // 8.25 KB (padded: bank-conflict-free A gather)
    __shared__ float s_logit[BT][4];
    __shared__ float s_attn[BT][4];

    const int tid   = threadIdx.x;
    const int bbase = blockIdx.x * BT;

    // ---- Stage 0: stage observation rows into LDS (skip leading 3 floats) ----
    for (int i = tid; i < BT * 96; i += 256) {
        int r = i / 96, c = i % 96;
        s_obs[r][c] = obs[(size_t)(bbase + r) * OBSD + 3 + c];
    }
    if (tid < BT * 4) s_logit[tid >> 2][tid & 3] = 0.0f;
    __syncthreads();

    // ---- Stage 1: per (batch,channel) graph-conv + logit accumulation ----
    // Layout in s_obs row: X[s][f]=ob[s*4+f]; A[t][s]=ob[16+t*4+s]; E[t][s][e]=ob[32+(t*4+s)*4+e]
    for (int p = 0; p < 8; ++p) {
        int task = p * 256 + tid;          // 0 .. 2047
        int b = task >> 7;                 // 0..15
        int c = task & 127;                // channel
        const float* ob = s_obs[b];

        float Wf[4][4], bf[4], Wr[4];
        #pragma unroll
        for (int e = 0; e < 4; ++e)
            #pragma unroll
            for (int f = 0; f < 4; ++f)
                Wf[e][f] = W_fgn[e * 512 + c * 4 + f];
        #pragma unroll
        for (int f = 0; f < 4; ++f) { bf[f] = b_fgn[c * 4 + f]; Wr[f] = W_root[f * 128 + c]; }

        // P[s][e] = sum_f Wf[e][f]*X[s][f] ; Q[s] = sum_f bf[f]*X[s][f]
        float P[4][4], Q[4];
        #pragma unroll
        for (int s = 0; s < 4; ++s) {
            Q[s] = 0.f;
            #pragma unroll
            for (int e = 0; e < 4; ++e) P[s][e] = 0.f;
            #pragma unroll
            for (int f = 0; f < 4; ++f) {
                float x = ob[s * 4 + f];
                Q[s] = fmaf(bf[f], x, Q[s]);
                #pragma unroll
                for (int e = 0; e < 4; ++e) P[s][e] = fmaf(Wf[e][f], x, P[s][e]);
            }
        }
        float bc = b_conv[c], akc = attn_k[c];
        #pragma unroll
        for (int t = 0; t < 4; ++t) {
            float msg = 0.f;
            #pragma unroll
            for (int s = 0; s < 4; ++s) {
                float inner = Q[s];
                #pragma unroll
                for (int e = 0; e < 4; ++e)
                    inner = fmaf(ob[32 + (t * 4 + s) * 4 + e], P[s][e], inner);
                msg = fmaf(ob[16 + t * 4 + s], inner, msg);
            }
            float rt = 0.f;
            #pragma unroll
            for (int f = 0; f < 4; ++f) rt = fmaf(ob[t * 4 + f], Wr[f], rt);
            float xc = msg + rt + bc;
            xc = xc > 0.f ? xc : 0.f;                 // relu
            s_xc[b][t][c] = xc;
            atomicAdd(&s_logit[b][t], xc * akc);      // ds_add_f32
        }
    }
    __syncthreads();

    // ---- Stage 2: 4-wide softmax per batch row ----
    if (tid < BT) {
        float l0 = s_logit[tid][0], l1 = s_logit[tid][1];
        float l2 = s_logit[tid][2], l3 = s_logit[tid][3];
        float m  = fmaxf(fmaxf(l0, l1), fmaxf(l2, l3));
        float e0 = expf(l0 - m), e1 = expf(l1 - m), e2 = expf(l2 - m), e3 = expf(l3 - m);
        float inv = 1.0f / (e0 + e1 + e2 + e3);
        s_attn[tid][0] = e0 * inv; s_attn[tid][1] = e1 * inv;
        s_attn[tid][2] = e2 * inv; s_attn[tid][3] = e3 * inv;
    }
    __syncthreads();

    // ---- Stage 3: attention pooling -> s_pool[16][128] ----
    for (int p = 0; p < 8; ++p) {
        int task = p * 256 + tid;
        int b = task >> 7, c = task & 127;
        float acc = 0.f;
        #pragma unroll
        for (int t = 0; t < 4; ++t) acc = fmaf(s_attn[b][t], s_xc[b][t][c], acc);
        s_pool[b][c] = acc;
    }
    __syncthreads();

    // ---- Stage 4: dense GEMM  pooled(16x128) @ W_dense(128x256), bias + tanh ----
    // Each of the 8 waves computes two 16x16 output tiles with V_WMMA_F32_16X16X4_F32.
    const int wid  = tid >> 5;
    const int lane = tid & 31;
    const int half = lane >> 4;     // K-pair select: lanes 0-15 -> K{0,1}, 16-31 -> K{2,3}
    const int mrow = lane & 15;     // A-matrix row / B-matrix column within tile

    #pragma unroll
    for (int ti = 0; ti < 2; ++ti) {
        const int tile = wid * 2 + ti;
        const int col  = tile * 16 + mrow;
        v8f acc = {};
        #pragma unroll 4
        for (int kk = 0; kk < NCH; kk += 4) {
            const int k0 = kk + half * 2;
            v2f a, bm;
            a.x  = s_pool[mrow][k0];
            a.y  = s_pool[mrow][k0 + 1];
            bm.x = W_dense[(size_t)k0 * NHID + col];
            bm.y = W_dense[(size_t)(k0 + 1) * NHID + col];
            acc = __builtin_amdgcn_wmma_f32_16x16x4_f32(
                /*neg_a=*/false, a, /*neg_b=*/false, bm,
                /*c_mod=*/(short)0, acc, /*reuse_a=*/false, /*reuse_b=*/false);
        }
        const float bd = b_dense[col];
        // D layout: VGPR r -> row (r + half*8), col = tile*16 + (lane&15)
        #pragma unroll
        for (int r = 0; r < 8; ++r) {
            int row = bbase + half * 8 + r;
            out[(size_t)row * NHID + col] = tanhf(acc[r] + bd);
        }
    }
}

extern "C" void kernel_launch(void* const* d_in, const int* in_sizes, int n_in,
                              void* d_out, int out_size, void* d_ws, size_t ws_size,
                              hipStream_t stream) {
    const float* obs     = (const float*)d_in[0];
    const float* W_fgn   = (const float*)d_in[1];
    const float* b_fgn   = (const float*)d_in[2];
    const float* W_root  = (const float*)d_in[3];
    const float* b_conv  = (const float*)d_in[4];
    const float* attn_k  = (const float*)d_in[5];
    const float* W_dense = (const float*)d_in[6];
    const float* b_dense = (const float*)d_in[7];
    float* out = (float*)d_out;

    const int B = in_sizes[0] / OBSD;      // 16384
    const int grid = B / BT;               // 1024 blocks of 256 threads (8 wave32)
    gnn_fused_kernel<<<grid, 256, 0, stream>>>(obs, W_fgn, b_fgn, W_root, b_conv,
                                               attn_k, W_dense, b_dense, out);
}